// ImprovedSSIUBlockV2_40407052321338
// MI455X (gfx1250) — compile-verified
//
#include <hip/hip_runtime.h>
#include <math.h>

#define B_ 8
#define C_ 64
#define H_ 256
#define W_ 256
#define HW_ (H_*W_)
#define EPSF 1e-5f
#define SGA_HID_ 16
#define CA_HID_ 8

typedef float v2f __attribute__((ext_vector_type(2)));
typedef float v8f __attribute__((ext_vector_type(8)));

__device__ __forceinline__ float sigm_(float x){ return 1.0f/(1.0f+expf(-x)); }
__device__ __forceinline__ float gelu_(float x){ return 0.5f*x*(1.0f+erff(x*0.70710678118654752f)); }

__device__ __forceinline__ float waveRed(float v){
    v += __shfl_xor(v, 16, 32);
    v += __shfl_xor(v,  8, 32);
    v += __shfl_xor(v,  4, 32);
    v += __shfl_xor(v,  2, 32);
    v += __shfl_xor(v,  1, 32);
    return v;
}

__device__ __forceinline__ v8f wmma_f32x4(v2f a, v2f b, v8f c){
    // D = A(16x4 f32) * B(4x16 f32) + C(16x16 f32)
    return __builtin_amdgcn_wmma_f32_16x16x4_f32(false, a, false, b, (short)0, c, false, false);
}

// ---------------- K1: LayerNorm1 + horizontal 1x13 dwconv ----------------
__global__ __launch_bounds__(256) void k1_ln_hconv(
    const float* __restrict__ x,
    const float* __restrict__ ln1w, const float* __restrict__ ln1b,
    const float* __restrict__ lk1w, const float* __restrict__ lk1b,
    float* __restrict__ h1, float* __restrict__ tbuf)
{
    __shared__ float sh[C_*W_];
    const int w = threadIdx.x;
    const int b = blockIdx.x / H_;
    const int h = blockIdx.x % H_;
    const size_t rowbase = ((size_t)(b*C_)*H_ + h)*(size_t)W_; // (b,0,h,0)

    float s = 0.f, s2 = 0.f;
    for (int c = 0; c < C_; ++c){
        float v = x[rowbase + (size_t)c*HW_ + w];
        sh[c*W_ + w] = v; s += v; s2 += v*v;
    }
    float mu = s * (1.0f/C_);
    float var = s2 * (1.0f/C_) - mu*mu;
    float rs = rsqrtf(var + EPSF);
    for (int c = 0; c < C_; ++c){
        float v = (sh[c*W_ + w] - mu)*rs*ln1w[c] + ln1b[c];
        sh[c*W_ + w] = v;
        h1[rowbase + (size_t)c*HW_ + w] = v;
    }
    __syncthreads();
    for (int c = 0; c < C_; ++c){
        float acc = lk1b[c];
        #pragma unroll
        for (int k = 0; k < 13; ++k){
            int ww = w + k - 6;
            if (ww >= 0 && ww < W_) acc += sh[c*W_ + ww] * lk1w[c*13 + k];
        }
        tbuf[rowbase + (size_t)c*HW_ + w] = acc;
    }
}

// ---------------- K2: vertical 13x1 + 3x3 + gate/proj 1x1 (WMMA fp32) + per-row gf partial ----------------
__global__ __launch_bounds__(256) void k2_salk(
    const float* __restrict__ h1g, const float* __restrict__ tg,
    const float* __restrict__ lk2w, const float* __restrict__ lk2b,
    const float* __restrict__ lk3w, const float* __restrict__ lk3b,
    const float* __restrict__ gatew, const float* __restrict__ gateb,
    const float* __restrict__ projw, const float* __restrict__ projb,
    float* __restrict__ salkg, float* __restrict__ rowsum)
{
    const int SP = W_ + 1;                 // padded stride (257) -> conflict-free channel-strided LDS access
    __shared__ float sh1[C_*W_];           // LayerNorm'd row (A operand for gate matmul)
    __shared__ float soc[C_*(W_+1)];       // dwconv output -> gated product -> salk (in place)
    __shared__ float sred[C_*8];           // per-wave partials (deterministic reduce)

    const int tid = threadIdx.x;
    const int w = tid;
    const int lane = tid & 31;
    const int wv = tid >> 5;
    const int b = blockIdx.x / H_;
    const int h = blockIdx.x % H_;
    const size_t rowbase = ((size_t)(b*C_)*H_ + h)*(size_t)W_;

    for (int c = 0; c < C_; ++c){
        const float* hc = h1g + rowbase + (size_t)c*HW_;  // (b,c,h,0)
        const float* tc = tg  + rowbase + (size_t)c*HW_;
        float hv = hc[w];
        sh1[c*W_ + w] = hv;
        float acc = lk2b[c] + lk3b[c];
        #pragma unroll
        for (int k = 0; k < 13; ++k){
            int hh = h + k - 6;
            if (hh >= 0 && hh < H_) acc += tc[(hh - h)*W_ + w] * lk2w[c*13 + k];
        }
        #pragma unroll
        for (int i = 0; i < 3; ++i){
            int hh = h + i - 1;
            if (hh < 0 || hh >= H_) continue;
            #pragma unroll
            for (int j = 0; j < 3; ++j){
                int ww = w + j - 1;
                if (ww < 0 || ww >= W_) continue;
                acc += hc[(hh - h)*W_ + ww] * lk3w[c*9 + i*3 + j];
            }
        }
        soc[c*SP + w] = acc;
    }
    __syncthreads();

    // WMMA fragment lane mapping (ISA 7.12.2, f32 16x16x4):
    //   A: lane<16 holds (M=lane, K=k0..k0+1); lane>=16 holds (M=lane-16, K=k0+2..k0+3)
    //   B: lane<16 holds (K=k0..k0+1, N=lane); lane>=16 holds (K=k0+2..k0+3, N=lane-16)
    //   D: lane<16 -> (M=v, N=lane); lane>=16 -> (M=v+8, N=lane-16)
    const int n = lane & 15;
    const int kk = (lane >> 4) * 2;
    const int mBase = (lane >> 4) * 8;
    const v8f zero8 = {0.f,0.f,0.f,0.f,0.f,0.f,0.f,0.f};

    for (int g = wv; g < 16; g += 8){
        const int w0 = g * 16;
        // ---- gate = sigmoid(W_g @ h1 + b_g), 16 pixels x 64 out-channels ----
        v8f acc[4] = {zero8, zero8, zero8, zero8};
        for (int k0 = 0; k0 < C_; k0 += 4){
            v2f a;
            a.x = sh1[(k0+kk  )*W_ + w0 + n];
            a.y = sh1[(k0+kk+1)*W_ + w0 + n];
            #pragma unroll
            for (int nt = 0; nt < 4; ++nt){
                int oc = nt*16 + n;
                v2f bb;
                bb.x = gatew[oc*C_ + k0 + kk];
                bb.y = gatew[oc*C_ + k0 + kk + 1];
                acc[nt] = wmma_f32x4(a, bb, acc[nt]);
            }
        }
        // gated product in place: soc[c][pixel] *= sigmoid(gate)
        #pragma unroll
        for (int nt = 0; nt < 4; ++nt){
            int oc = nt*16 + n;
            float gb = gateb[oc];
            #pragma unroll
            for (int v = 0; v < 8; ++v){
                int m = mBase + v;
                soc[oc*SP + w0 + m] *= sigm_(acc[nt][v] + gb);
            }
        }
        // ---- salk = W_p @ (out*gate) + b_p ----
        v8f pacc[4] = {zero8, zero8, zero8, zero8};
        for (int k0 = 0; k0 < C_; k0 += 4){
            v2f a;
            a.x = soc[(k0+kk  )*SP + w0 + n];
            a.y = soc[(k0+kk+1)*SP + w0 + n];
            #pragma unroll
            for (int nt = 0; nt < 4; ++nt){
                int oc = nt*16 + n;
                v2f bb;
                bb.x = projw[oc*C_ + k0 + kk];
                bb.y = projw[oc*C_ + k0 + kk + 1];
                pacc[nt] = wmma_f32x4(a, bb, pacc[nt]);
            }
        }
        #pragma unroll
        for (int nt = 0; nt < 4; ++nt){
            int oc = nt*16 + n;
            float pb = projb[oc];
            #pragma unroll
            for (int v = 0; v < 8; ++v){
                soc[oc*SP + w0 + mBase + v] = pacc[nt][v] + pb;
            }
        }
    }
    __syncthreads();

    // coalesced store + deterministic per-row reduction (wave tree + fixed-order partial sum)
    for (int c = 0; c < C_; ++c){
        float v = soc[c*SP + w];
        salkg[rowbase + (size_t)c*HW_ + w] = v;
        float s = waveRed(v);
        if (lane == 0) sred[c*8 + wv] = s;
    }
    __syncthreads();
    if (tid < C_){
        float s = 0.f;
        #pragma unroll
        for (int k = 0; k < 8; ++k) s += sred[tid*8 + k];
        rowsum[((size_t)(b*H_ + h))*C_ + tid] = s;
    }
}

// ---------------- KR: deterministic cross-row reduction: rowsum(B*H,C) -> sum(B,C) ----------------
__global__ void kr_reduce(const float* __restrict__ rowsum, float* __restrict__ outsum){
    int t = blockIdx.x*blockDim.x + threadIdx.x;
    if (t >= B_*C_) return;
    int b = t / C_, c = t % C_;
    float s = 0.f;
    for (int h = 0; h < H_; ++h)
        s += rowsum[((size_t)(b*H_ + h))*C_ + c];
    outsum[t] = s;
}

// ---------------- K3: SGA MLP -> g ----------------
__global__ void k3_g(const float* __restrict__ gfsum,
                     const float* __restrict__ w1, const float* __restrict__ b1,
                     const float* __restrict__ w2, const float* __restrict__ b2,
                     float* __restrict__ g)
{
    __shared__ float hid[B_*SGA_HID_];
    int t = threadIdx.x;
    if (t < B_*SGA_HID_){
        int bb = t / SGA_HID_, hh = t % SGA_HID_;
        float acc = b1[hh];
        for (int c = 0; c < C_; ++c)
            acc += (gfsum[bb*C_ + c] * (1.0f/HW_)) * w1[hh*C_ + c];
        hid[t] = gelu_(acc);
    }
    __syncthreads();
    if (t < B_*C_){
        int bb = t / C_, c = t % C_;
        float acc = b2[c];
        for (int k = 0; k < SGA_HID_; ++k)
            acc += hid[bb*SGA_HID_ + k] * w2[c*SGA_HID_ + k];
        g[t] = sigm_(acc);
    }
}

// ---------------- K4: x2 = x + g1*g*salk ; LN2 ; per-row y partial ; store x2 ----------------
__global__ __launch_bounds__(256) void k4_resid_ln2(
    const float* __restrict__ x, const float* __restrict__ salkg,
    const float* __restrict__ gamma1, const float* __restrict__ gv,
    const float* __restrict__ ln2w, const float* __restrict__ ln2b,
    float* __restrict__ x2g, float* __restrict__ rowsum)
{
    __shared__ float sh[C_*W_];
    __shared__ float sred[C_*8];
    const int tid = threadIdx.x;
    const int w = tid;
    const int lane = tid & 31;
    const int wv = tid >> 5;
    const int b = blockIdx.x / H_;
    const int h = blockIdx.x % H_;
    const size_t rowbase = ((size_t)(b*C_)*H_ + h)*(size_t)W_;

    float s = 0.f, s2 = 0.f;
    for (int c = 0; c < C_; ++c){
        float v = x[rowbase + (size_t)c*HW_ + w]
                + gamma1[c] * gv[b*C_ + c] * salkg[rowbase + (size_t)c*HW_ + w];
        sh[c*W_ + w] = v; s += v; s2 += v*v;
    }
    float mu = s * (1.0f/C_);
    float var = s2 * (1.0f/C_) - mu*mu;
    float rs = rsqrtf(var + EPSF);
    for (int c = 0; c < C_; ++c){
        float v = sh[c*W_ + w];
        x2g[rowbase + (size_t)c*HW_ + w] = v;
        float h2 = (v - mu)*rs*ln2w[c] + ln2b[c];
        float ss = waveRed(h2);
        if (lane == 0) sred[c*8 + wv] = ss;
    }
    __syncthreads();
    if (tid < C_){
        float ss = 0.f;
        #pragma unroll
        for (int k = 0; k < 8; ++k) ss += sred[tid*8 + k];
        rowsum[((size_t)(b*H_ + h))*C_ + tid] = ss;
    }
}

// ---------------- K5: channel-attention MLP -> a ----------------
__global__ void k5_a(const float* __restrict__ ysum,
                     const float* __restrict__ w1, const float* __restrict__ w2,
                     float* __restrict__ a)
{
    __shared__ float hid[B_*CA_HID_];
    int t = threadIdx.x;
    if (t < B_*CA_HID_){
        int bb = t / CA_HID_, hh = t % CA_HID_;
        float acc = 0.0f;
        for (int c = 0; c < C_; ++c)
            acc += (ysum[bb*C_ + c] * (1.0f/HW_)) * w1[hh*C_ + c];
        hid[t] = gelu_(acc);
    }
    __syncthreads();
    if (t < B_*C_){
        int bb = t / C_, c = t % C_;
        float acc = 0.0f;
        for (int k = 0; k < CA_HID_; ++k)
            acc += hid[bb*CA_HID_ + k] * w2[c*CA_HID_ + k];
        a[t] = sigm_(acc);
    }
}

// ---------------- K6: out = x2 + g2*a*LN2(x2) ----------------
__global__ __launch_bounds__(256) void k6_out(
    const float* __restrict__ x2g,
    const float* __restrict__ gamma2, const float* __restrict__ av,
    const float* __restrict__ ln2w, const float* __restrict__ ln2b,
    float* __restrict__ out)
{
    __shared__ float sh[C_*W_];
    const int w = threadIdx.x;
    const int b = blockIdx.x / H_;
    const int h = blockIdx.x % H_;
    const size_t rowbase = ((size_t)(b*C_)*H_ + h)*(size_t)W_;

    float s = 0.f, s2 = 0.f;
    for (int c = 0; c < C_; ++c){
        float v = x2g[rowbase + (size_t)c*HW_ + w];
        sh[c*W_ + w] = v; s += v; s2 += v*v;
    }
    float mu = s * (1.0f/C_);
    float var = s2 * (1.0f/C_) - mu*mu;
    float rs = rsqrtf(var + EPSF);
    for (int c = 0; c < C_; ++c){
        float v = sh[c*W_ + w];
        float h2 = (v - mu)*rs*ln2w[c] + ln2b[c];
        out[rowbase + (size_t)c*HW_ + w] = v + gamma2[c]*av[b*C_ + c]*h2;
    }
}

extern "C" void kernel_launch(void* const* d_in, const int* in_sizes, int n_in,
                              void* d_out, int out_size, void* d_ws, size_t ws_size,
                              hipStream_t stream)
{
    (void)in_sizes; (void)n_in; (void)out_size; (void)ws_size;
    const float* x     = (const float*)d_in[0];
    const float* ln1w  = (const float*)d_in[1];
    const float* ln1b  = (const float*)d_in[2];
    const float* ln2w  = (const float*)d_in[3];
    const float* ln2b  = (const float*)d_in[4];
    const float* gm1   = (const float*)d_in[5];
    const float* gm2   = (const float*)d_in[6];
    const float* lk1w  = (const float*)d_in[7];
    const float* lk1b  = (const float*)d_in[8];
    const float* lk2w  = (const float*)d_in[9];
    const float* lk2b  = (const float*)d_in[10];
    const float* lk3w  = (const float*)d_in[11];
    const float* lk3b  = (const float*)d_in[12];
    const float* projw = (const float*)d_in[13];
    const float* projb = (const float*)d_in[14];
    const float* gatew = (const float*)d_in[15];
    const float* gateb = (const float*)d_in[16];
    const float* sgaw1 = (const float*)d_in[17];
    const float* sgab1 = (const float*)d_in[18];
    const float* sgaw2 = (const float*)d_in[19];
    const float* sgab2 = (const float*)d_in[20];
    const float* caw1  = (const float*)d_in[21];
    const float* caw2  = (const float*)d_in[22];

    float* ws = (float*)d_ws;
    const size_t N  = (size_t)B_*C_*H_*W_;
    const size_t NR = (size_t)B_*H_*C_;     // per-row partial sums
    float* h1    = ws;                      // 128 MB
    float* tbuf  = ws + N;                  // 128 MB
    float* salk  = ws + 2*N;                // 128 MB
    float* rs1   = ws + 3*N;                // 512 KB
    float* rs2   = rs1 + NR;                // 512 KB
    float* gfsum = rs2 + NR;                // B*C
    float* ysum  = gfsum + B_*C_;
    float* g     = ysum  + B_*C_;
    float* a     = g     + B_*C_;
    float* x2    = h1;                      // h1 dead after k2 -> reuse
    float* out   = (float*)d_out;

    dim3 grid(B_*H_), blk(256);
    k1_ln_hconv <<<grid, blk, 0, stream>>>(x, ln1w, ln1b, lk1w, lk1b, h1, tbuf);
    k2_salk     <<<grid, blk, 0, stream>>>(h1, tbuf, lk2w, lk2b, lk3w, lk3b,
                                           gatew, gateb, projw, projb, salk, rs1);
    kr_reduce   <<<dim3(2), dim3(256), 0, stream>>>(rs1, gfsum);
    k3_g        <<<dim3(1), dim3(512), 0, stream>>>(gfsum, sgaw1, sgab1, sgaw2, sgab2, g);
    k4_resid_ln2<<<grid, blk, 0, stream>>>(x, salk, gm1, g, ln2w, ln2b, x2, rs2);
    kr_reduce   <<<dim3(2), dim3(256), 0, stream>>>(rs2, ysum);
    k5_a        <<<dim3(1), dim3(512), 0, stream>>>(ysum, caw1, caw2, a);
    k6_out      <<<grid, blk, 0, stream>>>(x2, gm2, a, ln2w, ln2b, out);
}